// Model03_54331336294859
// MI455X (gfx1250) — compile-verified
//
#include <hip/hip_runtime.h>
#include <stdint.h>

// ---------------- problem constants ----------------
#define T_LEN   2048
#define D_IN    8
#define CT      16                      // timesteps per TDM chunk
#define NCHUNK  (T_LEN / CT)            // 128
#define ROWDW   128                     // data DWORDs per LDS row (CT*8)
#define PADDW   6                       // TDM pad per row (pad_amount=5 -> 6 DW)
#define ROWF    (ROWDW + PADDW)         // 134 floats LDS row stride
#define BUFF    (32 * ROWF)             // 4288 floats per buffer
#define BUFBYTES (BUFF * 4)             // 17152 B

typedef float    v2f  __attribute__((ext_vector_type(2)));
typedef float    v8f  __attribute__((ext_vector_type(8)));
typedef unsigned u32x4 __attribute__((ext_vector_type(4)));
typedef int      i32x4 __attribute__((ext_vector_type(4)));
typedef int      i32x8 __attribute__((ext_vector_type(8)));

#if __has_builtin(__builtin_amdgcn_wmma_f32_16x16x4_f32)
#define HAVE_WMMA4 1
#else
#define HAVE_WMMA4 0
#endif

#if __has_builtin(__builtin_amdgcn_s_wait_tensorcnt)
#define WAIT_TENSOR(n) __builtin_amdgcn_s_wait_tensorcnt((short)(n))
#else
#define WAIT_TENSOR(n) asm volatile("s_wait_tensorcnt %0" ::"i"(n))
#endif

// Escape the LDS object into an opaque asm with a memory clobber: the compiler
// must then assume the asm (standing in for the TDM engine) wrote the shared
// array, so ds_loads from it cannot be undef-folded, hoisted above the
// tensorcnt wait, or sunk below the next tensor_load_to_lds issue.
#define LDS_SYNC(p) asm volatile("" ::"v"((const void*)(p)) : "memory")

template <bool V> struct BoolC { static constexpr bool value = V; };

static __device__ __forceinline__ float fast_sigmoid(float x) {
  // sigmoid(x) = 1 / (1 + 2^(-x*log2 e))   -> v_exp_f32 + v_rcp_f32
  float e = __builtin_amdgcn_exp2f(x * -1.44269504f);
  return __builtin_amdgcn_rcpf(1.0f + e);
}

static __device__ __forceinline__ float fast_tanh(float x) {
#if __has_builtin(__builtin_amdgcn_tanhf)
  return __builtin_amdgcn_tanhf(x);            // V_TANH_F32 on gfx1250
#else
  float e = __builtin_amdgcn_exp2f(x * 2.88539008f);   // exp(2x)
  return 1.0f - 2.0f * __builtin_amdgcn_rcpf(e + 1.0f);
#endif
}

// Issue one TDM 2D tile load: 32 rows (batch) x 128 floats (CT timesteps * 8),
// row stride = T*8 floats in memory, LDS rows padded to 134 DW (conflict-free).
static __device__ __forceinline__ void tdm_issue(const float* gptr, unsigned lds_byte_off) {
#if __has_builtin(__builtin_amdgcn_tensor_load_to_lds)
  unsigned long long ga = (unsigned long long)(uintptr_t)gptr;
  u32x4 g0;
  g0[0] = 1u;                                           // count=1, user desc
  g0[1] = lds_byte_off;                                 // lds_addr (bytes)
  g0[2] = (unsigned)(ga & 0xffffffffu);                 // global_addr[31:0]
  g0[3] = (unsigned)((ga >> 32) & 0x1ffffffu) | (2u << 30); // addr[56:32] | type=2
  i32x8 g1;
  g1[0] = (int)((2u << 16) |        // data_size = 4B
                (1u << 20) |        // pad_enable
                (6u << 22) |        // pad_interval: 128 DWORDs
                (5u << 25));        // pad_amount:   6 DWORDs
  g1[1] = (int)(16384u << 16);      // tensor_dim0[15:0] = T*8 = 16384
  g1[2] = (int)(4096u << 16);       // dim0[31:16]=0 | tensor_dim1[15:0]=4096
  g1[3] = (int)(128u << 16);        // dim1[31:16]=0 | tile_dim0 = 128
  g1[4] = 32;                       // tile_dim1 = 32 rows, tile_dim2 = 0
  g1[5] = 16384;                    // tensor_dim0_stride[31:0] = T*8
  g1[6] = 0;                        // stride0[47:32]=0 | dim1_stride[15:0]=0
  g1[7] = 0;
  i32x4 gz = {0, 0, 0, 0};
#if defined(__clang_major__) && (__clang_major__ >= 23)
  i32x8 gz8 = {0, 0, 0, 0, 0, 0, 0, 0};
  __builtin_amdgcn_tensor_load_to_lds(g0, g1, gz, gz, gz8, 0);
#else
  __builtin_amdgcn_tensor_load_to_lds(g0, g1, gz, gz, 0);
#endif
#else
  (void)gptr; (void)lds_byte_off;
#endif
}

__global__ __launch_bounds__(32, 1)
void gru6_fused_kernel(const float* __restrict__ x,
                       const float* __restrict__ Wih0,   // [3][8]
                       const float* __restrict__ Wrest,  // [5][3]
                       const float* __restrict__ Whh,    // [6][3]
                       const float* __restrict__ bih,    // [6][3]
                       const float* __restrict__ bhh,    // [6][3]
                       float* __restrict__ out) {
  __shared__ float smem[2 * BUFF];

  const int lane = threadIdx.x;
  const int m    = lane & 15;
  const int hi   = (lane >> 4) & 1;
  const int dsel = hi << 1;               // 0 or 2
  const int b0   = blockIdx.x * 32;

  // ---- small weights -> (uniform) registers; fold biases ----
  float WH[6][3], BH2[6], S0[6], S1[6], B2[6], WI[6][3];
#pragma unroll
  for (int l = 0; l < 6; ++l) {
    WH[l][0] = Whh[l * 3 + 0];
    WH[l][1] = Whh[l * 3 + 1];
    WH[l][2] = Whh[l * 3 + 2];
    S0[l] = bih[l * 3 + 0] + bhh[l * 3 + 0];   // r-gate bias sum
    S1[l] = bih[l * 3 + 1] + bhh[l * 3 + 1];   // z-gate bias sum
    B2[l] = bih[l * 3 + 2];                    // n-gate input bias
    BH2[l] = bhh[l * 3 + 2];                   // n-gate hidden bias (inside r*)
    WI[l][0] = WI[l][1] = WI[l][2] = 0.0f;
  }
#pragma unroll
  for (int l = 1; l < 6; ++l) {
    WI[l][0] = Wrest[(l - 1) * 3 + 0];
    WI[l][1] = Wrest[(l - 1) * 3 + 1];
    WI[l][2] = Wrest[(l - 1) * 3 + 2];
  }

#if HAVE_WMMA4
  // A-matrix (16x4 f32): M=gate rows (0..2 used), K split lanes 0-15 (K0,K1) / 16-31 (K2,K3)
  float a0x = 0.f, a0y = 0.f, a1x = 0.f, a1y = 0.f;
  if (m < 3) {
    const float* wr = Wih0 + m * 8 + dsel;
    a0x = wr[0]; a0y = wr[1];      // K = dsel, dsel+1
    a1x = wr[4]; a1y = wr[5];      // K = dsel+4, dsel+5
  }
  const v2f Alo = {a0x, a0y};
  const v2f Ahi = {a1x, a1y};
  // C accumulator preloaded with per-gate biases (rows 0..2)
  const v8f cbias = {S0[0], S1[0], B2[0], 0.f, 0.f, 0.f, 0.f, 0.f};
#else
  float W0[8], W1[8], W2[8];
#pragma unroll
  for (int d = 0; d < 8; ++d) {
    W0[d] = Wih0[d]; W1[d] = Wih0[8 + d]; W2[d] = Wih0[16 + d];
  }
#endif

  float h0 = 0.f, h1 = 0.f, h2 = 0.f, h3 = 0.f, h4 = 0.f, h5 = 0.f;

  const float* xblk = x + (size_t)b0 * T_LEN * D_IN;

  auto cell0 = [&](float xp0, float xp1, float xp2, float h) -> float {
    float r = fast_sigmoid(fmaf(WH[0][0], h, xp0));
    float z = fast_sigmoid(fmaf(WH[0][1], h, xp1));
    float n = fast_tanh(fmaf(r, fmaf(WH[0][2], h, BH2[0]), xp2));
    return fmaf(z, h - n, n);
  };
  auto cellN = [&](int l, float u, float h) -> float {
    float a0 = fmaf(WI[l][0], u, S0[l]);
    float r  = fast_sigmoid(fmaf(WH[l][0], h, a0));
    float a1 = fmaf(WI[l][1], u, S1[l]);
    float z  = fast_sigmoid(fmaf(WH[l][1], h, a1));
    float a2 = fmaf(WI[l][2], u, B2[l]);
    float n  = fast_tanh(fmaf(r, fmaf(WH[l][2], h, BH2[l]), a2));
    return fmaf(z, h - n, n);
  };

  // One chunk = CT timesteps; layers are skewed (layer l runs at t - l), so the
  // 6 cell updates per iteration are independent (ILP 6 within one lane).
  auto chunk_body = [&](auto first_tag, int base) {
    constexpr bool FIRST = decltype(first_tag)::value;
    const float* bp = &smem[base];
#pragma unroll
    for (int j = 0; j < CT; ++j) {
      float xp0, xp1, xp2;
#if HAVE_WMMA4
      const int lo = m * ROWF + j * 8 + dsel;
      v2f bAlo = *(const v2f*)(bp + lo);                  // batch rows 0..15
      v2f bAhi = *(const v2f*)(bp + lo + 4);
      v2f bBlo = *(const v2f*)(bp + lo + 16 * ROWF);      // batch rows 16..31
      v2f bBhi = *(const v2f*)(bp + lo + 16 * ROWF + 4);
      // D(gate m, batch n) = sum_d W[m][d] * x[n][d] + bias  (K=8 via 2 chained WMMAs)
      v8f accA = __builtin_amdgcn_wmma_f32_16x16x4_f32(false, Alo, false, bAlo, (short)0, cbias, false, false);
      accA     = __builtin_amdgcn_wmma_f32_16x16x4_f32(false, Ahi, false, bAhi, (short)0, accA,  false, false);
      v8f accB = __builtin_amdgcn_wmma_f32_16x16x4_f32(false, Alo, false, bBlo, (short)0, cbias, false, false);
      accB     = __builtin_amdgcn_wmma_f32_16x16x4_f32(false, Ahi, false, bBhi, (short)0, accB,  false, false);
      // gates for batch b0+n land in lane n, VGPR g; move upper-half results up 16 lanes
      float s0 = __shfl(accB[0], m, 32);
      float s1 = __shfl(accB[1], m, 32);
      float s2 = __shfl(accB[2], m, 32);
      xp0 = hi ? s0 : accA[0];
      xp1 = hi ? s1 : accA[1];
      xp2 = hi ? s2 : accA[2];
#else
      const int lo = lane * ROWF + j * 8;
      v2f p0 = *(const v2f*)(bp + lo + 0);
      v2f p1 = *(const v2f*)(bp + lo + 2);
      v2f p2 = *(const v2f*)(bp + lo + 4);
      v2f p3 = *(const v2f*)(bp + lo + 6);
      float xv[8] = {p0.x, p0.y, p1.x, p1.y, p2.x, p2.y, p3.x, p3.y};
      xp0 = S0[0]; xp1 = S1[0]; xp2 = B2[0];
#pragma unroll
      for (int d = 0; d < 8; ++d) {
        xp0 = fmaf(W0[d], xv[d], xp0);
        xp1 = fmaf(W1[d], xv[d], xp1);
        xp2 = fmaf(W2[d], xv[d], xp2);
      }
#endif
      // skewed pipeline: all cells read last-iteration state (independent)
      float h0n = cell0(xp0, xp1, xp2, h0);
      float h1n = cellN(1, h0, h1);
      float h2n = cellN(2, h1, h2);
      float h3n = cellN(3, h2, h3);
      float h4n = cellN(4, h3, h4);
      float h5n = cellN(5, h4, h5);
      h0 = h0n;
      if (!FIRST || j >= 1) h1 = h1n;
      if (!FIRST || j >= 2) h2 = h2n;
      if (!FIRST || j >= 3) h3 = h3n;
      if (!FIRST || j >= 4) h4 = h4n;
      if (!FIRST || j >= 5) h5 = h5n;
    }
  };

  // ---------------- TDM double-buffered main pipeline ----------------
  LDS_SYNC(smem);
  tdm_issue(xblk, 0);
  tdm_issue(xblk + (CT * D_IN), BUFBYTES);
  WAIT_TENSOR(1);                 // chunk 0 resident (TDM in-order per wave)
  LDS_SYNC(smem);
  chunk_body(BoolC<true>{}, 0);

  for (int c = 1; c < NCHUNK; ++c) {
    LDS_SYNC(smem);               // keep prior chunk's ds reads above next issue
    if (c + 1 < NCHUNK) {
      tdm_issue(xblk + (size_t)(c + 1) * (CT * D_IN), (unsigned)(((c + 1) & 1) * BUFBYTES));
      WAIT_TENSOR(1);             // chunk c resident
    } else {
      WAIT_TENSOR(0);
    }
    LDS_SYNC(smem);               // keep this chunk's ds reads below the wait
    chunk_body(BoolC<false>{}, (c & 1) * BUFF);
  }

  // ---------------- flush skewed pipeline (5 iterations) ----------------
#pragma unroll
  for (int k = 0; k < 5; ++k) {
    float h1n = cellN(1, h0, h1);
    float h2n = cellN(2, h1, h2);
    float h3n = cellN(3, h2, h3);
    float h4n = cellN(4, h3, h4);
    float h5n = cellN(5, h4, h5);
    h1 = h1n; h2 = h2n; h3 = h3n; h4 = h4n; h5 = h5n;
  }

  out[b0 + lane] = h5;
}

extern "C" void kernel_launch(void* const* d_in, const int* in_sizes, int n_in,
                              void* d_out, int out_size, void* d_ws, size_t ws_size,
                              hipStream_t stream) {
  (void)n_in; (void)d_ws; (void)ws_size; (void)out_size;
  const float* x     = (const float*)d_in[0];
  const float* Wih0  = (const float*)d_in[1];
  const float* Wrest = (const float*)d_in[2];
  const float* Whh   = (const float*)d_in[3];
  const float* bih   = (const float*)d_in[4];
  const float* bhh   = (const float*)d_in[5];
  float* out = (float*)d_out;

  const int Bsz    = in_sizes[0] / (T_LEN * D_IN);   // 4096
  const int blocks = Bsz / 32;                       // 128
  gru6_fused_kernel<<<dim3(blocks), dim3(32), 0, stream>>>(x, Wih0, Wrest, Whh, bih, bhh, out);
}